// TopologicalHallucinationEnergy_1580547975311
// MI455X (gfx1250) — compile-verified
//
#include <hip/hip_runtime.h>
#include <hip/hip_fp16.h>

typedef float    v8f  __attribute__((ext_vector_type(8)));
typedef float    v2f  __attribute__((ext_vector_type(2)));
typedef _Float16 h8   __attribute__((ext_vector_type(8)));

#define NPTS   8192
#define EPS_   1e-5f
#define BLUR_  0.01f
#define FI_    (1.0f / 1.01f)   // reg_m / (reg_m + reg)
#define NITER_ 200

// ---------------------------------------------------------------------------
// prep: topo = exp(-(b-d)^2/(2*sigma^2)), row/col squared norms, init u,v,cmax
// ---------------------------------------------------------------------------
__global__ void prep_kernel(const float* __restrict__ b, const float* __restrict__ d,
                            const float* __restrict__ p, const float* __restrict__ g,
                            float* __restrict__ topo, float* __restrict__ psq,
                            float* __restrict__ gsq, float* __restrict__ u,
                            float* __restrict__ v, unsigned* __restrict__ cmax_bits) {
    int i = blockIdx.x * blockDim.x + threadIdx.x;
    if (i < NPTS) {
        float pers = b[i] - d[i];
        topo[i] = __expf(-pers * pers * 50.0f);     // 1/(2*0.1^2) = 50
        float x = p[3*i], y = p[3*i+1], z = p[3*i+2];
        psq[i] = x*x + y*y + z*z;
        x = g[3*i]; y = g[3*i+1]; z = g[3*i+2];
        gsq[i] = x*x + y*y + z*z;
        u[i] = 1.0f / NPTS;
        v[i] = 1.0f / NPTS;
    }
    if (i == 0) *cmax_bits = 0u;   // C >= 0, so uint-ordered float max works
}

// ---------------------------------------------------------------------------
// C_max pass: WMMA f32 16x16x4 computes p@g^T tiles (K=3 padded to 4).
// Each block owns one 16-row tile; 8 waves sweep the 512 col tiles.
// Operand ternaries select on the loaded VALUE (branch-free v_cndmask),
// never on the address, so the inner loop has no exec-mask branches.
// ---------------------------------------------------------------------------
__global__ __launch_bounds__(256)
void cmax_kernel(const float* __restrict__ p, const float* __restrict__ g,
                 const float* __restrict__ topo, const float* __restrict__ psq,
                 const float* __restrict__ gsq, unsigned* __restrict__ cmax_bits) {
    const int lane = threadIdx.x & 31;
    const int wave = threadIdx.x >> 5;
    const int hi   = lane >> 4;        // K-half: lanes 0-15 -> K=0,1 ; 16-31 -> K=2,3
    const int l    = lane & 15;
    const int k0   = hi * 2;           // 0 or 2
    const int rowbase = blockIdx.x * 16;

    // A tile (16x4 f32): lane holds A[l][k0], A[l][k0+1]; K=3 padded with 0
    int ar = rowbase + l;
    float ax = p[3*ar + k0];
    float ay = p[3*ar + 1];            // always in bounds; value-selected below
    v2f a;
    a.x = ax;
    a.y = hi ? 0.0f : ay;

    // row factors for the 8 accumulator regs: row = rowbase + r + hi*8
    float tp[8], ps[8];
#pragma unroll
    for (int r = 0; r < 8; ++r) {
        int row = rowbase + r + hi * 8;
        tp[r] = topo[row]; ps[r] = psq[row];
    }

    float mx = 0.0f;
    for (int j = wave; j < 512; j += 8) {
        int bc = j * 16 + l;
        float bx = g[3*bc + k0];
        float by = g[3*bc + 1];
        v2f bm;                         // B tile (4x16): B[k0][l], B[k0+1][l]
        bm.x = bx;
        bm.y = hi ? 0.0f : by;
        v8f c = {};
        c = __builtin_amdgcn_wmma_f32_16x16x4_f32(false, a, false, bm,
                                                  (short)0, c, false, false);
        float gq = gsq[bc];             // col = j*16 + l for every r
#pragma unroll
        for (int r = 0; r < 8; ++r) {
            float sp = fmaxf(ps[r] + gq - 2.0f * c[r], 0.0f);
            mx = fmaxf(mx, tp[r] * sp);
        }
    }
    for (int off = 16; off > 0; off >>= 1) mx = fmaxf(mx, __shfl_xor(mx, off, 32));
    if (lane == 0) atomicMax(cmax_bits, __float_as_uint(mx));
}

// ---------------------------------------------------------------------------
// Build K = exp(-C / (cmax+eps) / blur) in fp16 (128 MB -> resident in 192MB L2)
// Each thread produces 8 consecutive columns -> one coalesced 16B store.
// ---------------------------------------------------------------------------
__global__ __launch_bounds__(256)
void buildK_kernel(const float* __restrict__ p, const float* __restrict__ g,
                   const float* __restrict__ topo, const float* __restrict__ psq,
                   const float* __restrict__ gsq, const unsigned* __restrict__ cmax_bits,
                   _Float16* __restrict__ K) {
    long gid  = (long)blockIdx.x * blockDim.x + threadIdx.x;
    long base = gid * 8;
    int row = (int)(base >> 13);
    int col = (int)(base & (NPTS - 1));
    float cmax  = __uint_as_float(*cmax_bits) + EPS_;
    float scale = -1.0f / (cmax * BLUR_);
    float px = p[3*row], py = p[3*row+1], pz = p[3*row+2];
    float ps = psq[row], tp = topo[row];
    h8 out;
#pragma unroll
    for (int k = 0; k < 8; ++k) {
        int c = col + k;
        float dot = px * g[3*c] + py * g[3*c+1] + pz * g[3*c+2];
        float sp  = fmaxf(ps + gsq[c] - 2.0f * dot, 0.0f);
        out[k] = (_Float16)__expf(tp * sp * scale);
    }
    *(h8*)(K + base) = out;
}

// ---------------------------------------------------------------------------
// Async VGPR-bypassing stage of a 32KB fp32 vector (8192 floats) into LDS:
// 8 x GLOBAL_LOAD_ASYNC_TO_LDS_B128 per thread (ASYNCcnt-tracked). The
// instruction offset is added to BOTH the global and the LDS address (ISA
// CDNA5 ISA), so one address setup serves all 8 transfers.
// LDS byte offset = low 32 bits of the flat shared pointer (ISA 10.2).
// ---------------------------------------------------------------------------
__device__ __forceinline__ void async_stage_vec(float* lds_dst, const float* src,
                                                unsigned tid) {
    unsigned ldsa = (unsigned)(size_t)lds_dst + tid * 16u;
    unsigned goff = tid * 16u;
    asm volatile(
        "global_load_async_to_lds_b128 %0, %1, %2 offset:0\n\t"
        "global_load_async_to_lds_b128 %0, %1, %2 offset:4096\n\t"
        "global_load_async_to_lds_b128 %0, %1, %2 offset:8192\n\t"
        "global_load_async_to_lds_b128 %0, %1, %2 offset:12288\n\t"
        "global_load_async_to_lds_b128 %0, %1, %2 offset:16384\n\t"
        "global_load_async_to_lds_b128 %0, %1, %2 offset:20480\n\t"
        "global_load_async_to_lds_b128 %0, %1, %2 offset:24576\n\t"
        "global_load_async_to_lds_b128 %0, %1, %2 offset:28672\n\t"
        :: "v"(ldsa), "v"(goff), "s"(src) : "memory");
    asm volatile("s_wait_asynccnt 0x0" ::: "memory");
}

// ---------------------------------------------------------------------------
// u = (a / (K v))^fi     — one wave per row, v async-staged in LDS
// ---------------------------------------------------------------------------
__global__ __launch_bounds__(256)
void sinkhorn_rows(const _Float16* __restrict__ K, const float* __restrict__ v,
                   const float* __restrict__ a, float* __restrict__ u) {
    __shared__ float vs[NPTS];
    async_stage_vec(vs, v, threadIdx.x);
    __syncthreads();
    int lane = threadIdx.x & 31;
    int wave = threadIdx.x >> 5;
    int row  = blockIdx.x * 8 + wave;
    const _Float16* Kr = K + (long)row * NPTS;
    float acc = 0.0f;
    for (int it = 0; it < 32; ++it) {
        int b0 = it * 256 + lane * 8;
        __builtin_prefetch(Kr + b0 + 2048, 0, 1);   // global_prefetch, 4KB ahead
        h8 k8 = *(const h8*)(Kr + b0);
#pragma unroll
        for (int q = 0; q < 8; ++q) acc += (float)k8[q] * vs[b0 + q];
    }
    for (int off = 16; off > 0; off >>= 1) acc += __shfl_xor(acc, off, 32);
    if (lane == 0) u[row] = __powf(a[row] / acc, FI_);
}

// ---------------------------------------------------------------------------
// v = (b / (K^T u))^fi   — block owns 64 cols; u async-staged in LDS;
// 4 row-slices + LDS reduce
// ---------------------------------------------------------------------------
__global__ __launch_bounds__(256)
void sinkhorn_cols(const _Float16* __restrict__ K, const float* __restrict__ u,
                   const float* __restrict__ b, float* __restrict__ v) {
    __shared__ float us[NPTS];
    __shared__ float red[256];
    async_stage_vec(us, u, threadIdx.x);
    __syncthreads();
    int c   = threadIdx.x & 63;
    int rg  = threadIdx.x >> 6;          // 0..3 row-slice
    int col = blockIdx.x * 64 + c;
    float acc = 0.0f;
    for (int n = rg; n < NPTS; n += 4)
        acc += (float)K[(long)n * NPTS + col] * us[n];
    red[threadIdx.x] = acc;
    __syncthreads();
    if (rg == 0) {
        float t = red[c] + red[c + 64] + red[c + 128] + red[c + 192];
        v[col] = __powf(b[col] / t, FI_);
    }
}

// ---------------------------------------------------------------------------
// Final: sum u_n * K_nm * C_nm * v_m  (== the_val * C_max), WMMA-tiled C
// ---------------------------------------------------------------------------
__global__ __launch_bounds__(256)
void final_kernel(const float* __restrict__ p, const float* __restrict__ g,
                  const float* __restrict__ topo, const float* __restrict__ psq,
                  const float* __restrict__ gsq, const _Float16* __restrict__ K,
                  const float* __restrict__ u, const float* __restrict__ v,
                  float* __restrict__ partials) {
    __shared__ float red[8];
    const int lane = threadIdx.x & 31;
    const int wave = threadIdx.x >> 5;
    const int hi   = lane >> 4;
    const int l    = lane & 15;
    const int k0   = hi * 2;
    const int rowbase = blockIdx.x * 16;

    int ar = rowbase + l;
    float ax = p[3*ar + k0];
    float ay = p[3*ar + 1];
    v2f a;
    a.x = ax;
    a.y = hi ? 0.0f : ay;

    float tp[8], ps[8], uu[8];
    int rows[8];
#pragma unroll
    for (int r = 0; r < 8; ++r) {
        int row = rowbase + r + hi * 8;
        rows[r] = row; tp[r] = topo[row]; ps[r] = psq[row]; uu[r] = u[row];
    }

    float acc = 0.0f;
    for (int j = wave; j < 512; j += 8) {
        int bc = j * 16 + l;
        float bx = g[3*bc + k0];
        float by = g[3*bc + 1];
        v2f bm;
        bm.x = bx;
        bm.y = hi ? 0.0f : by;
        v8f c = {};
        c = __builtin_amdgcn_wmma_f32_16x16x4_f32(false, a, false, bm,
                                                  (short)0, c, false, false);
        float gq = gsq[bc], vv = v[bc];
#pragma unroll
        for (int r = 0; r < 8; ++r) {
            float sp  = fmaxf(ps[r] + gq - 2.0f * c[r], 0.0f);
            float Cnm = tp[r] * sp;
            float kk  = (float)K[(long)rows[r] * NPTS + bc];
            acc += uu[r] * vv * kk * Cnm;
        }
    }
    for (int off = 16; off > 0; off >>= 1) acc += __shfl_xor(acc, off, 32);
    if (lane == 0) red[wave] = acc;
    __syncthreads();
    if (threadIdx.x == 0) {
        float s = 0.0f;
        for (int w = 0; w < 8; ++w) s += red[w];
        partials[blockIdx.x] = s;
    }
}

__global__ void reduce_kernel(const float* __restrict__ partials, float* __restrict__ out) {
    __shared__ float sh[512];
    sh[threadIdx.x] = partials[threadIdx.x];
    __syncthreads();
    for (int s = 256; s > 0; s >>= 1) {
        if (threadIdx.x < s) sh[threadIdx.x] += sh[threadIdx.x + s];
        __syncthreads();
    }
    if (threadIdx.x == 0) out[0] = sh[0];
}

// ---------------------------------------------------------------------------
extern "C" void kernel_launch(void* const* d_in, const int* in_sizes, int n_in,
                              void* d_out, int out_size, void* d_ws, size_t ws_size,
                              hipStream_t stream) {
    const float* b_vals = (const float*)d_in[0];
    const float* d_vals = (const float*)d_in[1];
    const float* p      = (const float*)d_in[2];   // [N,3]
    const float* g      = (const float*)d_in[3];   // [M,3]
    const float* mass_p = (const float*)d_in[4];   // a
    const float* mass_g = (const float*)d_in[5];   // b

    // workspace layout: 128MB fp16 K (fits the 192MB L2) + small vectors
    char* ws = (char*)d_ws;
    _Float16* K = (_Float16*)ws;
    size_t off = (size_t)NPTS * NPTS * sizeof(_Float16);
    float*    u    = (float*)(ws + off);  off += NPTS * sizeof(float);
    float*    v    = (float*)(ws + off);  off += NPTS * sizeof(float);
    float*    topo = (float*)(ws + off);  off += NPTS * sizeof(float);
    float*    psq  = (float*)(ws + off);  off += NPTS * sizeof(float);
    float*    gsq  = (float*)(ws + off);  off += NPTS * sizeof(float);
    unsigned* cmax = (unsigned*)(ws + off); off += 16;
    float*    partials = (float*)(ws + off);

    prep_kernel<<<NPTS / 256, 256, 0, stream>>>(b_vals, d_vals, p, g,
                                                topo, psq, gsq, u, v, cmax);
    cmax_kernel<<<512, 256, 0, stream>>>(p, g, topo, psq, gsq, cmax);
    buildK_kernel<<<(NPTS * (NPTS / 8)) / 256, 256, 0, stream>>>(p, g, topo, psq, gsq,
                                                                 cmax, K);
    for (int it = 0; it < NITER_; ++it) {
        sinkhorn_rows<<<NPTS / 8, 256, 0, stream>>>(K, v, mass_p, u);
        sinkhorn_cols<<<NPTS / 64, 256, 0, stream>>>(K, u, mass_g, v);
    }
    final_kernel<<<512, 256, 0, stream>>>(p, g, topo, psq, gsq, K, u, v, partials);
    reduce_kernel<<<1, 512, 0, stream>>>(partials, (float*)d_out);
}